// AffineRegistration_15006615733257
// MI455X (gfx1250) — compile-verified
//
#include <hip/hip_runtime.h>
#include <math.h>

// Problem constants (reference: D,H,W = 128,192,192; 10 iters; LR=1; EPS=1e-8)
#define DD 128
#define HH 192
#define WW 192
#define NVOX (DD*HH*WW)
#define NBLK 512
#define NTHR 256
#define NQ   40          // 36 gram sums + S1,S2,S3 + pad
#define NQP  48          // padded partials stride (3 WMMA batches of 16, no guards)
#define WS_PARAMS 0      // 16 floats: theta (4x4)
#define WS_MEANF  16
#define WS_TVAR   17
#define WS_PART   64     // NBLK*NQP floats of block partials

typedef float v2f __attribute__((ext_vector_type(2)));
typedef float v8f __attribute__((ext_vector_type(8)));

__device__ __forceinline__ float waveSum(float v){
  #pragma unroll
  for (int off = 16; off > 0; off >>= 1) v += __shfl_xor(v, off, 32);
  return v;
}

__device__ __forceinline__ float tapLoad(const float* __restrict__ img, int x, int y, int z){
  bool ok = ((unsigned)x < (unsigned)WW) && ((unsigned)y < (unsigned)HH) && ((unsigned)z < (unsigned)DD);
  int xc = min(max(x, 0), WW - 1);
  int yc = min(max(y, 0), HH - 1);
  int zc = min(max(z, 0), DD - 1);
  float v = img[(zc * HH + yc) * WW + xc];
  return ok ? v : 0.0f;
}

struct SampleOut { float p, sx, sy, sz; };

// Trilinear sample + analytic spatial derivatives scaled to grid coords
__device__ __forceinline__ SampleOut trisample(const float* __restrict__ img,
                                               float gx, float gy, float gz){
  float fx = ((gx + 1.0f) * (float)WW - 1.0f) * 0.5f;
  float fy = ((gy + 1.0f) * (float)HH - 1.0f) * 0.5f;
  float fz = ((gz + 1.0f) * (float)DD - 1.0f) * 0.5f;
  float x0f = floorf(fx), y0f = floorf(fy), z0f = floorf(fz);
  float wx = fx - x0f, wy = fy - y0f, wz = fz - z0f;
  int x0 = (int)x0f, y0 = (int)y0f, z0 = (int)z0f;
  float v000 = tapLoad(img, x0,   y0,   z0  );
  float v001 = tapLoad(img, x0+1, y0,   z0  );
  float v010 = tapLoad(img, x0,   y0+1, z0  );
  float v011 = tapLoad(img, x0+1, y0+1, z0  );
  float v100 = tapLoad(img, x0,   y0,   z0+1);
  float v101 = tapLoad(img, x0+1, y0,   z0+1);
  float v110 = tapLoad(img, x0,   y0+1, z0+1);
  float v111 = tapLoad(img, x0+1, y0+1, z0+1);
  float cx00 = v000 + wx * (v001 - v000);
  float cx01 = v010 + wx * (v011 - v010);
  float cx10 = v100 + wx * (v101 - v100);
  float cx11 = v110 + wx * (v111 - v110);
  float cy0  = cx00 + wy * (cx01 - cx00);
  float cy1  = cx10 + wy * (cx11 - cx10);
  SampleOut o;
  o.p = cy0 + wz * (cy1 - cy0);
  float d00 = v001 - v000, d01 = v011 - v010, d10 = v101 - v100, d11 = v111 - v110;
  float dy0 = d00 + wy * (d01 - d00);
  float dy1 = d10 + wy * (d11 - d10);
  float dfx = dy0 + wz * (dy1 - dy0);
  float e0  = cx01 - cx00, e1 = cx11 - cx10;
  float dfy = e0 + wz * (e1 - e0);
  float dfz = cy1 - cy0;
  o.sx = dfx * (0.5f * (float)WW);
  o.sy = dfy * (0.5f * (float)HH);
  o.sz = dfz * (0.5f * (float)DD);
  return o;
}

__global__ void ar_initKernel(float* __restrict__ ws){
  int t = threadIdx.x;
  if (t < 16) ws[WS_PARAMS + t] = ((t % 5) == 0) ? 1.0f : 0.0f;  // 4x4 identity
}

// Fixed-image stats: block partials of (sum f, sum f^2)
__global__ void ar_fstatKernel(const float* __restrict__ fixedv, float* __restrict__ ws){
  int tid = blockIdx.x * blockDim.x + threadIdx.x;
  float s1 = 0.0f, s2 = 0.0f;
  for (int i = tid; i < NVOX; i += NBLK * NTHR){
    float f = fixedv[i];
    s1 += f; s2 += f * f;
  }
  s1 = waveSum(s1); s2 = waveSum(s2);
  __shared__ float l1[8], l2[8];
  int lane = threadIdx.x & 31, wave = threadIdx.x >> 5;
  if (lane == 0){ l1[wave] = s1; l2[wave] = s2; }
  __syncthreads();
  if (threadIdx.x == 0){
    float a = 0.0f, b = 0.0f;
    #pragma unroll
    for (int w = 0; w < 8; ++w){ a += l1[w]; b += l2[w]; }
    ws[WS_PART + blockIdx.x * NQP + 0] = a;
    ws[WS_PART + blockIdx.x * NQP + 1] = b;
  }
}

__global__ void ar_ffinKernel(float* __restrict__ ws){
  float s1 = 0.0f, s2 = 0.0f;
  for (int b = threadIdx.x; b < NBLK; b += NTHR){
    s1 += ws[WS_PART + b * NQP + 0];
    s2 += ws[WS_PART + b * NQP + 1];
  }
  s1 = waveSum(s1); s2 = waveSum(s2);
  __shared__ float l1[8], l2[8];
  int lane = threadIdx.x & 31, wave = threadIdx.x >> 5;
  if (lane == 0){ l1[wave] = s1; l2[wave] = s2; }
  __syncthreads();
  if (threadIdx.x == 0){
    float a = 0.0f, b = 0.0f;
    #pragma unroll
    for (int w = 0; w < 8; ++w){ a += l1[w]; b += l2[w]; }
    float n = (float)NVOX;
    ws[WS_MEANF] = a / n;               // mean(fixed)
    ws[WS_TVAR]  = b - a * a / n;       // T = sum(tc^2)
  }
}

// Main pass: per-voxel sample + 39 partial sums per block.
// q layout: 0..11 = sum(tc*s_r*b_c); 12..23 = sum(p*s_r*b_c); 24..35 = sum(s_r*b_c)
//           36 = sum p; 37 = sum p^2; 38 = sum p*tc; 39..47 = zero pad
__global__ void ar_passKernel(const float* __restrict__ moving,
                              const float* __restrict__ fixedv,
                              float* __restrict__ ws){
  const float t00 = ws[0],  t01 = ws[1],  t02 = ws[2],  t03 = ws[3];
  const float t10 = ws[4],  t11 = ws[5],  t12 = ws[6],  t13 = ws[7];
  const float t20 = ws[8],  t21 = ws[9],  t22 = ws[10], t23 = ws[11];
  const float meanF = ws[WS_MEANF];
  float acc[NQ];
  #pragma unroll
  for (int q = 0; q < NQ; ++q) acc[q] = 0.0f;

  int tid = blockIdx.x * blockDim.x + threadIdx.x;
  for (int i = tid; i < NVOX; i += NBLK * NTHR){
    int x = i % WW;
    int t = i / WW;
    int y = t % HH;
    int z = t / HH;
    float bx = (2.0f * (float)x + 1.0f) / (float)WW - 1.0f;
    float by = (2.0f * (float)y + 1.0f) / (float)HH - 1.0f;
    float bz = (2.0f * (float)z + 1.0f) / (float)DD - 1.0f;
    float gx = t00 * bx + t01 * by + t02 * bz + t03;
    float gy = t10 * bx + t11 * by + t12 * bz + t13;
    float gz = t20 * bx + t21 * by + t22 * bz + t23;
    SampleOut s = trisample(moving, gx, gy, gz);
    float tc = fixedv[i] - meanF;
    float p  = s.p;
    float s3v[3] = { s.sx, s.sy, s.sz };
    float u3v[3] = { tc, p, 1.0f };
    float b4v[4] = { bx, by, bz, 1.0f };
    #pragma unroll
    for (int ui = 0; ui < 3; ++ui){
      #pragma unroll
      for (int ri = 0; ri < 3; ++ri){
        float wv = u3v[ui] * s3v[ri];
        #pragma unroll
        for (int ci = 0; ci < 4; ++ci)
          acc[ui * 12 + ri * 4 + ci] += wv * b4v[ci];
      }
    }
    acc[36] += p;
    acc[37] += p * p;
    acc[38] += p * tc;
  }

  __shared__ float red[8][NQ];
  int lane = threadIdx.x & 31, wave = threadIdx.x >> 5;
  #pragma unroll
  for (int q = 0; q < NQ; ++q){
    float v = waveSum(acc[q]);
    if (lane == 0) red[wave][q] = v;
  }
  __syncthreads();
  if (threadIdx.x < NQP){
    int q = threadIdx.x;
    float v = 0.0f;
    if (q < NQ){
      #pragma unroll
      for (int w = 0; w < 8; ++w) v += red[w][q];
    }
    ws[WS_PART + blockIdx.x * NQP + q] = v;   // q >= NQ written as 0 (pad)
  }
}

// WMMA reduction of NBLKx48 partials: A(16 quantities x 4 blocks) * ones(4x16)
// accumulated into C; then finalize gradient / sim. Waves 0..2 take batches 0/16/32.
// Padded stride makes every load in the WMMA loop unconditional (no exec masking).
__global__ void ar_reduceKernel(float* __restrict__ ws, float* __restrict__ dout, int mode){
  __shared__ float tot[48];
  const float* partials = ws + WS_PART;
  int lane = threadIdx.x & 31, wave = threadIdx.x >> 5;

  if (wave < 3){
    int q  = wave * 16 + (lane & 15);   // always < 48: loads are in-bounds, pad is zero
    int hi = lane >> 4;                 // 0: lanes 0-15 (K=0,1) ; 1: lanes 16-31 (K=2,3)
    v8f c = {};
    v2f b; b[0] = 1.0f; b[1] = 1.0f;    // all-ones B: D[m,n] = sum_k A[m,k]
    for (int ch = 0; ch < NBLK / 4; ++ch){
      int blk = ch * 4 + hi * 2;
      v2f a;
      a[0] = partials[blk       * NQP + q];
      a[1] = partials[(blk + 1) * NQP + q];
      c = __builtin_amdgcn_wmma_f32_16x16x4_f32(false, a, false, b,
                                                (short)0, c, false, false);
    }
    // C/D layout: VGPR j -> M=j (lanes 0-15), M=j+8 (lanes 16-31); value same for all N
    if ((lane & 15) == 0){
      int mbase = hi * 8;
      #pragma unroll
      for (int j = 0; j < 8; ++j){
        int qq = wave * 16 + mbase + j;
        if (qq < NQ) tot[qq] = c[j];
      }
    }
  }
  __syncthreads();

  if (threadIdx.x == 0){
    float n  = (float)NVOX;
    float S1 = tot[36], S2 = tot[37], S3 = tot[38];
    float meanP = S1 / n;
    float P  = S2 - S1 * S1 / n;        // sum(pc^2)
    float T  = ws[WS_TVAR];             // sum(tc^2)
    float N  = S3;                      // sum(pc*tc), sum(tc)==0
    float sq  = sqrtf(P * T);
    float den = sq + 1e-8f;
    if (mode == 0){
      float alpha = -1.0f / den;
      float beta  = N * T / (den * den * sq);
      #pragma unroll
      for (int r = 0; r < 3; ++r){
        #pragma unroll
        for (int cc = 0; cc < 4; ++cc){
          int q = r * 4 + cc;
          float g = alpha * tot[q] + beta * (tot[12 + q] - meanP * tot[24 + q]);
          ws[WS_PARAMS + q] -= 1.0f * g;   // LR = 1.0
        }
      }
    } else {
      dout[NVOX + 16] = -N / den;          // sim
    }
  }
  if (mode == 1 && threadIdx.x < 16)
    dout[NVOX + threadIdx.x] = ws[WS_PARAMS + threadIdx.x];
}

__global__ void ar_outKernel(const float* __restrict__ moving,
                             const float* __restrict__ ws,
                             float* __restrict__ dout){
  const float t00 = ws[0],  t01 = ws[1],  t02 = ws[2],  t03 = ws[3];
  const float t10 = ws[4],  t11 = ws[5],  t12 = ws[6],  t13 = ws[7];
  const float t20 = ws[8],  t21 = ws[9],  t22 = ws[10], t23 = ws[11];
  int tid = blockIdx.x * blockDim.x + threadIdx.x;
  for (int i = tid; i < NVOX; i += NBLK * NTHR){
    int x = i % WW;
    int t = i / WW;
    int y = t % HH;
    int z = t / HH;
    float bx = (2.0f * (float)x + 1.0f) / (float)WW - 1.0f;
    float by = (2.0f * (float)y + 1.0f) / (float)HH - 1.0f;
    float bz = (2.0f * (float)z + 1.0f) / (float)DD - 1.0f;
    float gx = t00 * bx + t01 * by + t02 * bz + t03;
    float gy = t10 * bx + t11 * by + t12 * bz + t13;
    float gz = t20 * bx + t21 * by + t22 * bz + t23;
    SampleOut s = trisample(moving, gx, gy, gz);
    dout[i] = s.p;
  }
}

extern "C" void kernel_launch(void* const* d_in, const int* in_sizes, int n_in,
                              void* d_out, int out_size, void* d_ws, size_t ws_size,
                              hipStream_t stream) {
  const float* moving = (const float*)d_in[0];
  const float* fixedv = (const float*)d_in[1];
  float* out = (float*)d_out;
  float* ws  = (float*)d_ws;

  ar_initKernel<<<1, 32, 0, stream>>>(ws);
  ar_fstatKernel<<<NBLK, NTHR, 0, stream>>>(fixedv, ws);
  ar_ffinKernel<<<1, NTHR, 0, stream>>>(ws);
  for (int it = 0; it < 10; ++it){
    ar_passKernel<<<NBLK, NTHR, 0, stream>>>(moving, fixedv, ws);
    ar_reduceKernel<<<1, 128, 0, stream>>>(ws, out, 0);
  }
  ar_passKernel<<<NBLK, NTHR, 0, stream>>>(moving, fixedv, ws);   // stats at final params
  ar_reduceKernel<<<1, 128, 0, stream>>>(ws, out, 1);             // sim + params -> d_out
  ar_outKernel<<<NBLK, NTHR, 0, stream>>>(moving, ws, out);       // warped volume -> d_out
}